// SkipGram_3874060501389
// MI455X (gfx1250) — compile-verified
//
#include <hip/hip_runtime.h>
#include <math.h>

// ---------------------------------------------------------------------------
// SkipGram negative-sampling loss, MI455X (gfx1250, wave32).
//
// Roofline: 983,040 gathered rows x 512B ~= 503 MB random HBM gather out of a
// 512MB table (> 192MB L2) => ~22us at 23.3 TB/s. Compute is ~0.25 GFLOP, so
// the kernel is pure bandwidth: coalesced 512B-row gathers, f32 end to end.
// The per-phrase dot-with-u is done with V_WMMA_F32_16X16X4_F32 (16 phrases
// per wave, K accumulated over 32 chunks of 4) — free at this intensity and
// keeps full f32 precision for the fragile (score > 0) mask.
// ---------------------------------------------------------------------------

typedef __attribute__((ext_vector_type(2))) float v2f;
typedef __attribute__((ext_vector_type(4))) float v4f;
typedef __attribute__((ext_vector_type(8))) float v8f;

#define DIM        128
#define LTOK       5
#define PPOS       32768
#define NNEG       (PPOS * 5)            // 163840
#define NPHRASE    (PPOS + NNEG)         // 196608
#define WAVES_PB   4
#define TILE       16                    // phrases per wave
#define PHRASES_PB (WAVES_PB * TILE)     // 64
#define NBLOCKS    (NPHRASE / PHRASES_PB)// 3072
#define POSBLOCKS  (PPOS / PHRASES_PB)   // 512
#define ROWPAD     132                   // 128 + 4 pad floats: row r's b64 A-
                                         // fragment read lands on bank 4r+...,
                                         // all 64 banks covered, conflict-free

// ---------------------------------------------------------------------------
// Kernel 1: u = mean over the 5 pos_u token rows (includes the 1/5 factor).
// ---------------------------------------------------------------------------
__global__ void sg_build_u(const int* __restrict__ pos_u,
                           const float* __restrict__ emb,
                           float* __restrict__ u_out) {
    int i = threadIdx.x;                  // 0..127
    float s = 0.f;
    for (int t = 0; t < LTOK; ++t)
        s += emb[(size_t)pos_u[t] * DIM + i];
    u_out[i] = 0.2f * s;
}

// ---------------------------------------------------------------------------
// Kernel 2: per-wave 16-phrase tile: coalesced gather+mean into LDS, then
// scores via WMMA f32 16x16x4, nonlinearity, fixed-order block partial.
// Blocks [0, POSBLOCKS) are entirely positive phrases; the rest negative.
// ---------------------------------------------------------------------------
__global__ __launch_bounds__(128) void sg_score(
    const int* __restrict__ pos_v, const int* __restrict__ neg_v,
    const float* __restrict__ emb, const float* __restrict__ u_g,
    float* __restrict__ partials) {

    __shared__ __align__(16) float tile[WAVES_PB][TILE][ROWPAD];
    __shared__ float u_sh[DIM];
    __shared__ float wave_part[WAVES_PB * 2];

    const int tid  = threadIdx.x;
    const int wave = tid >> 5;
    const int lane = tid & 31;
    const int half = lane >> 4;           // 0 or 1 (K half of fragment)
    const int m    = lane & 15;           // row within 16x16 fragment

    // stage u (all 128 threads, one element each)
    u_sh[tid] = u_g[tid];

    // ---- gather + mean: one 512B row per global_load_b128 across the wave
    const int g0 = (blockIdx.x * WAVES_PB + wave) * TILE;   // first phrase id
    for (int p = 0; p < TILE; ++p) {
        const int g = g0 + p;
        const int* ip = (g < PPOS) ? (pos_v + (size_t)g * LTOK)
                                   : (neg_v + (size_t)(g - PPOS) * LTOK);
        v4f acc = {0.f, 0.f, 0.f, 0.f};
        for (int t = 0; t < LTOK; ++t) {
            const v4f* src = (const v4f*)(emb + (size_t)ip[t] * DIM) + lane;
            acc += *src;
        }
        acc *= 0.2f;                                     // EmbeddingBag mean
        *(v4f*)&tile[wave][p][lane * 4] = acc;
    }
    __syncthreads();   // publish LDS tile + u_sh across lanes

    // ---- scores: D(16x16) += A(16x4) x B(4x16), 32 chunks covering K=128.
    // A layout (ISA 7.12.2, 32-bit A 16x4): lane L holds row M=L&15,
    //   VGPR0 = K = 4c + 2*(L>>4), VGPR1 = K+1.
    // B is u broadcast into all 16 columns, so every column of D is the
    // phrase score; assumed B K-row mapping mirrors A's K split (if the HW
    // maps K rows 0,1|2,3 instead of 0,2|1,3 the broadcast indices change,
    // but columns stay identical — only the K pairing assumption matters).
    v8f d = {0.f, 0.f, 0.f, 0.f, 0.f, 0.f, 0.f, 0.f};
    for (int c = 0; c < 32; ++c) {
        const int k = 4 * c + 2 * half;
        v2f a = *(const v2f*)&tile[wave][m][k];
        v2f b;
        b.x = u_sh[k];
        b.y = u_sh[k + 1];
        d = __builtin_amdgcn_wmma_f32_16x16x4_f32(
                /*neg_a=*/false, a, /*neg_b=*/false, b,
                /*c_mod=*/(short)0, d, /*reuse_a=*/false, /*reuse_b=*/false);
    }

    // ---- nonlinearity + partial sums.
    // D layout: lane L, VGPR r = element (M = r + 8*(L>>4), N = L&15);
    // all N identical, so lanes with m==0 own phrases r + 8*half.
    if (m == 0) {
        const bool isneg = (g0 >= PPOS);   // tiles never straddle pos/neg
        float part = 0.f;
        for (int r = 0; r < 8; ++r) {
            float s = d[r];
            if (isneg) {
                float t = s * (1.0f / (float)NNEG);
                part += (t > 0.f) ? expf(t) : 0.f;
            } else {
                part += s;
            }
        }
        wave_part[wave * 2 + half] = part;
    }
    __syncthreads();

    if (tid == 0) {                        // fixed-order => deterministic
        float s = 0.f;
        for (int i = 0; i < WAVES_PB * 2; ++i) s += wave_part[i];
        partials[blockIdx.x] = s;
    }
}

// ---------------------------------------------------------------------------
// Kernel 3: deterministic serial reduction of 3072 block partials -> loss.
// ---------------------------------------------------------------------------
__global__ void sg_finalize(const float* __restrict__ partials,
                            float* __restrict__ out) {
    if (threadIdx.x == 0) {
        float pos = 0.f, neg = 0.f;
        for (int i = 0; i < POSBLOCKS; ++i)          pos += partials[i];
        for (int i = POSBLOCKS; i < NBLOCKS; ++i)    neg += partials[i];
        float pos_loss = pos / (float)PPOS;
        float neg_loss = -logf(1.0f + neg);
        out[0] = -(neg_loss + pos_loss);
    }
}

// ---------------------------------------------------------------------------
extern "C" void kernel_launch(void* const* d_in, const int* in_sizes, int n_in,
                              void* d_out, int out_size, void* d_ws, size_t ws_size,
                              hipStream_t stream) {
    const int*   pos_u = (const int*)d_in[0];     // [5]
    const int*   pos_v = (const int*)d_in[1];     // [P,5]
    const int*   neg_v = (const int*)d_in[2];     // [N,5]
    const float* emb   = (const float*)d_in[3];   // [1e6,128]
    float*       out   = (float*)d_out;           // scalar

    float* ws       = (float*)d_ws;
    float* u        = ws;            // 128 floats
    float* partials = ws + 128;      // NBLOCKS floats

    sg_build_u<<<1, DIM, 0, stream>>>(pos_u, emb, u);
    sg_score<<<NBLOCKS, 128, 0, stream>>>(pos_v, neg_v, emb, u, partials);
    sg_finalize<<<1, 32, 0, stream>>>(partials, out);
}